// CustomNetwork_37031208026716
// MI455X (gfx1250) — compile-verified
//
#include <hip/hip_runtime.h>
#include <hip/hip_bf16.h>
#include <stdint.h>

#define WIDTH   512
#define HALF    256
#define DEPTH   32
#define BATCH   65536

#define RP 4                        // row-pairs per lane (float2 over rows)
#define ROWS_PER_WAVE  (2*RP)       // 8
#define NWAVES 8
#define ROWS_PER_BLOCK (NWAVES*ROWS_PER_WAVE)   // 64
#define PSTRIDE 8                   // floats per (layer,slot) param record (32B)
#define PLAYER  (WIDTH*PSTRIDE)     // 4096 floats = 16KB per layer

typedef float v2f __attribute__((ext_vector_type(2)));
typedef float v4f __attribute__((ext_vector_type(4)));
typedef int   v4i __attribute__((ext_vector_type(4)));

// ---------------- LDS staging of one layer's params (async DMA path) --------
__device__ __forceinline__ void stage_layer(const float* __restrict__ PP, int l,
                                            float* dst, int tid) {
    const v4f* __restrict__ src = (const v4f*)(PP + (size_t)l * PLAYER);
    v4f* d4 = (v4f*)dst;
#pragma unroll
    for (int i = 0; i < 4; ++i) {
        int k = tid + i * 256;                 // 1024 x 16B = 16KB per layer
        // global -> LDS async DMA, tracked by ASYNCcnt (gfx1250)
        __builtin_amdgcn_global_load_async_to_lds_b128(
            (v4i*)(src + k), (v4i*)(d4 + k), 0, 0);
    }
}

__device__ __forceinline__ void wait_async_all() {
    __builtin_amdgcn_s_wait_asynccnt(0);
}

// ---------------- per-element nonlinearity (packed f32) ---------------------
__device__ __forceinline__ v2f nonlin(v2f n, v4f p, v2f pb) {
    v2f u = n * p.z - pb.x;            // v_pk_fma_f32
    v2f t = u * u + pb.y;              // v_pk_fma_f32
    v2f rt;
    rt.x = __builtin_amdgcn_sqrtf(t.x);
    rt.y = __builtin_amdgcn_sqrtf(t.y);
    return u + p.w * rt;               // v_pk_fma_f32
}

// intra-thread butterfly: partner slot differs in register bit RB
template <int RB>
__device__ __forceinline__ void butter(v2f (&d)[16][RP], const float* P, int lane) {
#pragma unroll
    for (int r = 0; r < 16; ++r) {
        if (r & RB) continue;
        const int pr = r | RB;
        const int phi0 = (r << 5) | lane;
        const int phi1 = (pr << 5) | lane;
        const v4f p0  = *(const v4f*)(P + phi0 * PSTRIDE);
        const v2f p0b = *(const v2f*)(P + phi0 * PSTRIDE + 4);
        const v4f p1  = *(const v4f*)(P + phi1 * PSTRIDE);
        const v2f p1b = *(const v2f*)(P + phi1 * PSTRIDE + 4);
#pragma unroll
        for (int q = 0; q < RP; ++q) {
            v2f x0 = d[r][q];
            v2f x1 = d[pr][q];
            v2f n0 = x0 * p0.x + x1 * p0.y;   // c*x0 + s*x1
            v2f n1 = x1 * p1.x + x0 * p1.y;   // c*x1 - s*x0 (ps pre-negated)
            d[r][q]  = nonlin(n0, p0, p0b);
            d[pr][q] = nonlin(n1, p1, p1b);
        }
    }
}

// cross-lane butterfly: partner differs in lane bit (mask)
__device__ __forceinline__ void shuffle_layer(v2f (&d)[16][RP], const float* P,
                                              int lane, int mask) {
#pragma unroll
    for (int r = 0; r < 16; ++r) {
        const int phi = (r << 5) | lane;
        const v4f p  = *(const v4f*)(P + phi * PSTRIDE);
        const v2f pb = *(const v2f*)(P + phi * PSTRIDE + 4);
#pragma unroll
        for (int q = 0; q < RP; ++q) {
            v2f x = d[r][q];
            v2f xp;
            xp.x = __shfl_xor(x.x, mask, 32);
            xp.y = __shfl_xor(x.y, mask, 32);
            v2f n = x * p.x + xp * p.y;       // ps sign encodes which side we are
            d[r][q] = nonlin(n, p, pb);
        }
    }
}

// ---------------- main kernel: whole network, rows resident in registers ----
// Slot phi = (r<<5)|lane holds logical feature rotl9(phi, l) at layer l.
// Layer l is an XOR-butterfly on slot bit b=(8-l)%9: bits 8..5 are register
// bits (intra-thread), bits 4..0 are lane bits (one __shfl_xor per element).
__global__ void __launch_bounds__(256)
net_kernel(const float* __restrict__ X, const float* __restrict__ PP,
           float* __restrict__ out) {
    __shared__ __align__(16) float pbuf[2][PLAYER];   // 32KB double buffer
    const int tid  = threadIdx.x;
    const int lane = tid & 31;
    const int wave = tid >> 5;
    const int row0 = blockIdx.x * ROWS_PER_BLOCK + wave * ROWS_PER_WAVE;

    // kick off layer-0 param DMA before anything else (max latency overlap)
    stage_layer(PP, 0, pbuf[0], tid);

    // load rows: slot phi=(r<<5)|lane <-> feature phi (identity at layer 0)
    v2f d[16][RP];
#pragma unroll
    for (int r = 0; r < 16; ++r) {
        const int col = (r << 5) | lane;
#pragma unroll
        for (int q = 0; q < RP; ++q) {
            d[r][q].x = X[(size_t)(row0 + 2 * q)     * WIDTH + col];
            d[r][q].y = X[(size_t)(row0 + 2 * q + 1) * WIDTH + col];
        }
    }

    wait_async_all();
    __syncthreads();

    for (int l = 0; l < DEPTH; ++l) {
        if (l + 1 < DEPTH) stage_layer(PP, l + 1, pbuf[(l + 1) & 1], tid);
        const float* P = pbuf[l & 1];
        const int b = 8 - (l % 9);          // butterfly bit of slot index
        switch (b) {
            case 8: butter<8>(d, P, lane); break;
            case 7: butter<4>(d, P, lane); break;
            case 6: butter<2>(d, P, lane); break;
            case 5: butter<1>(d, P, lane); break;
            default: shuffle_layer(d, P, lane, 1 << b); break;
        }
        wait_async_all();
        __syncthreads();
    }

    // final logical index = rotl9(phi, 32 mod 9 = 5)
#pragma unroll
    for (int r = 0; r < 16; ++r) {
        const int oidx = ((lane & 15) << 5) | (r << 1) | (lane >> 4);
#pragma unroll
        for (int q = 0; q < RP; ++q) {
            out[(size_t)(row0 + 2 * q)     * WIDTH + oidx] = d[r][q].x;
            out[(size_t)(row0 + 2 * q + 1) * WIDTH + oidx] = d[r][q].y;
        }
    }
}

// ---------------- prologue: fold all transcendentals into a param table -----
// Record PP[l][phi][8] = { c, ps, a, cc, b, invk, 0, 0 }
//   slot phi at layer l holds logical feature j = rotl9(phi, l)
//   rotation pair index p = j & 255, side = j>>8  (ps = +s or -s)
//   nonlinearity params indexed at g = rotl9(phi, l+1)
__device__ __forceinline__ int rot9(int x, int k) {
    return ((x << k) | (x >> (9 - k))) & 511;   // k in [0,8]; k==0 -> x
}

__global__ void param_kernel(const float* __restrict__ th,
                             const float* __restrict__ bias,
                             const float* __restrict__ s1,
                             const float* __restrict__ s2,
                             const float* __restrict__ curv,
                             float* __restrict__ PP) {
    int idx = blockIdx.x * blockDim.x + threadIdx.x;
    if (idx >= DEPTH * WIDTH) return;
    int l = idx >> 9;
    int phi = idx & 511;

    int j = rot9(phi, l % 9);
    int p = j & 255;
    int top = j >> 8;
    float c = cosf(th[l * HALF + p]);
    float s = sinf(th[l * HALF + p]);
    float ps = top ? -s : s;

    int g = rot9(phi, (l + 1) % 9);
    float m1 = expf(s1[l * WIDTH + g]);
    float m2 = expf(s2[l * WIDTH + g]);
    float a  = 0.5f * (m1 + m2);
    float cc = (m2 - m1) / (m1 + m2);
    float bb = sinhf(bias[l * WIDTH + g]);
    float invk = expf(-curv[l * WIDTH + g]);

    float* o = PP + (size_t)idx * PSTRIDE;
    o[0] = c;  o[1] = ps;   o[2] = a;   o[3] = cc;
    o[4] = bb; o[5] = invk; o[6] = 0.f; o[7] = 0.f;
}

// ---------------- launcher --------------------------------------------------
extern "C" void kernel_launch(void* const* d_in, const int* in_sizes, int n_in,
                              void* d_out, int out_size, void* d_ws, size_t ws_size,
                              hipStream_t stream) {
    const float* X    = (const float*)d_in[0];
    const float* th   = (const float*)d_in[1];
    const float* bias = (const float*)d_in[2];
    const float* s1   = (const float*)d_in[3];
    const float* s2   = (const float*)d_in[4];
    const float* cv   = (const float*)d_in[5];
    float* PP   = (float*)d_ws;              // needs DEPTH*WIDTH*8*4 = 512KB
    float* outp = (float*)d_out;

    param_kernel<<<(DEPTH * WIDTH + 255) / 256, 256, 0, stream>>>(th, bias, s1, s2, cv, PP);
    net_kernel<<<BATCH / ROWS_PER_BLOCK, 256, 0, stream>>>(X, PP, outp);
}